// GATLayer_regular_64003602645265
// MI455X (gfx1250) — compile-verified
//
#include <hip/hip_runtime.h>
#include <hip/hip_bf16.h>

#define N_NODES_C 100000
#define D_IN_C    128
#define D_OUT_C   64

typedef __attribute__((ext_vector_type(2))) float v2f;
typedef __attribute__((ext_vector_type(8))) float v8f;

// ---------------------------------------------------------------------------
// Kernel 1: fused dual-GEMM (x0@W1, x0@W2) + bias + ReLU + attention logits.
// One wave = one 16-row tile. V_WMMA_F32_16X16X4_F32, K stepped by 4, 32 steps.
// A-fragment (16x4 f32): lane%16 = M row, (lane/16)*2 = K pair base.
// B-fragment (4x16 f32): lane%16 = N col, (lane/16)*2 = K pair base.
// C/D (16x16 f32): lane%16 = N, VGPR v -> M = v + 8*(lane/16).
//
// Weights staged in LDS in K-pair-major float2 layout:
//   sW[p*64 + col] = { W[2p][col], W[2p+1][col] }
// so each B fragment is ONE aligned ds_load_b64 straight into the contiguous
// even-aligned VGPR pair WMMA SRC1 needs (no v_mov repacking).
// ---------------------------------------------------------------------------
__global__ __launch_bounds__(256) void gat_gemm_att_kernel(
    const float* __restrict__ x0,
    const float* __restrict__ W1, const float* __restrict__ b1,
    const float* __restrict__ W2, const float* __restrict__ b2,
    const float* __restrict__ wa1, const float* __restrict__ ba1,
    const float* __restrict__ wa2, const float* __restrict__ ba2,
    float* __restrict__ out,   // d_out: initialized with x0_i (agg added later)
    float* __restrict__ xj,    // ws: relu(x0@W2+b2)   [n, 64]
    float* __restrict__ a1,    // ws: x0_i@wa1 + ba1   [n]
    float* __restrict__ a2)    // ws: x0_j@wa2 + ba2   [n]
{
    // 64 K-pairs x 64 cols, float2 per slot -> 32 KB per matrix.
    __shared__ v2f sW1[(D_IN_C / 2) * D_OUT_C];
    __shared__ v2f sW2[(D_IN_C / 2) * D_OUT_C];

    const int t = threadIdx.x;

    // Stage + swizzle both weight matrices into K-pair-major LDS layout.
    // 4096 slots per matrix, 256 threads -> 16 slots each.
#pragma unroll
    for (int i = 0; i < (D_IN_C / 2) * D_OUT_C / 256; ++i) {   // 16 iters
        const int s   = i * 256 + t;
        const int kp  = s >> 6;        // K pair
        const int col = s & 63;
        v2f w;
        w.x = W1[(2 * kp)     * D_OUT_C + col];
        w.y = W1[(2 * kp + 1) * D_OUT_C + col];
        sW1[s] = w;                    // ds_store_b64
        w.x = W2[(2 * kp)     * D_OUT_C + col];
        w.y = W2[(2 * kp + 1) * D_OUT_C + col];
        sW2[s] = w;
    }
    __syncthreads();

    const int wave = t >> 5;        // 0..7
    const int lane = t & 31;
    const int half = lane >> 4;     // K-pair / M-half selector
    const int l16  = lane & 15;

    const int row_base = (blockIdx.x * 8 + wave) * 16;
    int arow = row_base + l16;
    if (arow >= N_NODES_C) arow = N_NODES_C - 1;   // clamp tail loads (stores guarded)

    v8f acc1[4] = {};   // 4 N-tiles of x0@W1
    v8f acc2[4] = {};   // 4 N-tiles of x0@W2

    const float* aptr = x0 + (size_t)arow * D_IN_C + half * 2;
    // Per-lane LDS base: column l16, K-pair offset "half"; all 256 fragment
    // loads become immediate offsets from these two base addresses.
    const v2f* __restrict__ w1base = sW1 + (size_t)half * D_OUT_C + l16;
    const v2f* __restrict__ w2base = sW2 + (size_t)half * D_OUT_C + l16;

#pragma unroll
    for (int k0 = 0; k0 < D_IN_C; k0 += 4) {
        v2f a;
        a.x = aptr[k0];         // merges into global_load_b64
        a.y = aptr[k0 + 1];
        const int kpbase = (k0 >> 1) * D_OUT_C;   // K-pair row in LDS
#pragma unroll
        for (int nt = 0; nt < 4; ++nt) {
            const int idx = kpbase + nt * 16;
            v2f bfrag = w1base[idx];              // ds_load_b64, aligned pair
            acc1[nt] = __builtin_amdgcn_wmma_f32_16x16x4_f32(
                false, a, false, bfrag, (short)0, acc1[nt], false, false);
            bfrag = w2base[idx];
            acc2[nt] = __builtin_amdgcn_wmma_f32_16x16x4_f32(
                false, a, false, bfrag, (short)0, acc2[nt], false, false);
        }
    }

    const float bias_a1 = ba1[0];
    const float bias_a2 = ba2[0];

    // Epilogue: bias + ReLU, store x0_i -> out, x0_j -> ws, reduce logits.
#pragma unroll
    for (int v = 0; v < 8; ++v) {
        const int m = v + half * 8;
        const int row = row_base + m;
        if (row < N_NODES_C) {
            float p1 = 0.f, p2 = 0.f;
#pragma unroll
            for (int nt = 0; nt < 4; ++nt) {
                const int n = nt * 16 + l16;
                float h1 = acc1[nt][v] + b1[n];
                h1 = h1 > 0.f ? h1 : 0.f;
                float h2 = acc2[nt][v] + b2[n];
                h2 = h2 > 0.f ? h2 : 0.f;
                out[(size_t)row * D_OUT_C + n] = h1;   // x0_i (agg atomically added later)
                xj[(size_t)row * D_OUT_C + n]  = h2;   // x0_j
                p1 += h1 * wa1[n];
                p2 += h2 * wa2[n];
            }
            // 16-lane tree reduction within this half (row is uniform per half).
#pragma unroll
            for (int s = 8; s >= 1; s >>= 1) {
                p1 += __shfl_xor(p1, s, 16);
                p2 += __shfl_xor(p2, s, 16);
            }
            if (l16 == 0) {
                a1[row] = p1 + bias_a1;
                a2[row] = p2 + bias_a2;
            }
        }
    }
}

// ---------------------------------------------------------------------------
// Kernel 2: edge-wise sigmoid attention + scatter-add. 16 lanes per edge,
// float4 gather of x0_j[dst] (coalesced 256B per edge), f32 atomics into out.
// x0_j (25.6 MB) and out (25.6 MB) are L2-resident (192 MB L2).
// ---------------------------------------------------------------------------
__global__ __launch_bounds__(256) void gat_edge_kernel(
    const long long* __restrict__ ei,   // [2, E] int64
    const float* __restrict__ a1,
    const float* __restrict__ a2,
    const float* __restrict__ xj,
    float* __restrict__ out,
    int n_edges)
{
    const int tid = blockIdx.x * 256 + threadIdx.x;
    const int e = tid >> 4;
    if (e >= n_edges) return;
    const int c = (tid & 15) * 4;

    const long long src = ei[e];
    const long long dst = ei[(size_t)n_edges + e];

    const float s = a1[src] + a2[dst];
    const float att = 1.0f / (1.0f + __expf(-s));

    const float4 xv = *(const float4*)(xj + (size_t)dst * D_OUT_C + c);
    float* op = out + (size_t)src * D_OUT_C + c;
    atomicAdd(op + 0, att * xv.x);
    atomicAdd(op + 1, att * xv.y);
    atomicAdd(op + 2, att * xv.z);
    atomicAdd(op + 3, att * xv.w);
}

extern "C" void kernel_launch(void* const* d_in, const int* in_sizes, int n_in,
                              void* d_out, int out_size, void* d_ws, size_t ws_size,
                              hipStream_t stream) {
    (void)n_in; (void)out_size; (void)ws_size;
    const float*     x0  = (const float*)d_in[0];
    /* d_in[1] (x1) unused by the module */
    const long long* ei  = (const long long*)d_in[2];
    const float*     W1  = (const float*)d_in[3];
    const float*     b1  = (const float*)d_in[4];
    const float*     W2  = (const float*)d_in[5];
    const float*     b2  = (const float*)d_in[6];
    const float*     wa1 = (const float*)d_in[7];
    const float*     ba1 = (const float*)d_in[8];
    const float*     wa2 = (const float*)d_in[9];
    const float*     ba2 = (const float*)d_in[10];

    float* out = (float*)d_out;
    float* ws  = (float*)d_ws;
    float* xj  = ws;                                   // [n, 64]
    float* a1  = ws + (size_t)N_NODES_C * D_OUT_C;     // [n]
    float* a2  = a1 + N_NODES_C;                       // [n]

    const int n_edges = in_sizes[2] / 2;

    // Kernel 1: 8 waves x 16 rows = 128 rows per block.
    const int rows_per_block = 128;
    const int grid1 = (N_NODES_C + rows_per_block - 1) / rows_per_block;
    gat_gemm_att_kernel<<<grid1, 256, 0, stream>>>(
        x0, W1, b1, W2, b2, wa1, ba1, wa2, ba2, out, xj, a1, a2);

    // Kernel 2: 16 threads per edge.
    const long long total_threads = (long long)n_edges * 16;
    const int grid2 = (int)((total_threads + 255) / 256);
    gat_edge_kernel<<<grid2, 256, 0, stream>>>(ei, a1, a2, xj, out, n_edges);
}